// Block_72464688218281
// MI455X (gfx1250) — compile-verified
//
#include <hip/hip_runtime.h>
#include <hip/hip_bf16.h>

typedef __attribute__((ext_vector_type(2))) float v2f;
typedef __attribute__((ext_vector_type(8))) float v8f;

#define B_SZ   4
#define N_PTS  40960
#define K_NB   16
#define D_OUT  16
#define BN_EPS 1e-5f

// One wave32 per point (b,n). Lanes 0..15 / 16..31 both build the 10-channel
// positional feature for neighbor k = lane&15, then feed 3 chained
// V_WMMA_F32_16X16X4_F32 ops (channels 0-3, 4-7, 8-9+pad) to produce the
// 16x16 (d_out x K) tile, followed by fused eval-BN + ReLU and coalesced
// non-temporal stores. Native v_sqrt_f32 / v_rsq_f32 via builtins (no libm
// denorm-rescue ladders; operands here are >= 0 / >= eps).
__global__ __launch_bounds__(256) void relpos_enc_wmma_kernel(
    const float* __restrict__ xyz,          // [B, N, 3]
    const long long* __restrict__ neigh_idx,// [B, N, K] int64
    const float* __restrict__ Wmat,         // [D_OUT, 10]
    const float* __restrict__ gamma,        // [D_OUT]
    const float* __restrict__ beta,         // [D_OUT]
    const float* __restrict__ rmean,        // [D_OUT]
    const float* __restrict__ rvar,         // [D_OUT]
    float* __restrict__ out)                // [B, D_OUT, N, K]
{
    const int wave = (blockIdx.x << 3) + (threadIdx.x >> 5);
    const int lane = threadIdx.x & 31;
    const int b    = wave >> 15;            // wave / N_PTS (N_PTS = 2^15 * 1.25... no:
    // N_PTS = 40960 is not a power of two; keep exact div below.
    const int bq   = wave / N_PTS;
    const int n    = wave - bq * N_PTS;
    const int k    = lane & 15;
    const bool hi  = lane >= 16;
    (void)b;

    // Center ("tile") point — uniform across the wave, broadcast via cache.
    const float* xb = xyz + (size_t)bq * N_PTS * 3;
    const float tx = xb[n * 3 + 0];
    const float ty = xb[n * 3 + 1];
    const float tz = xb[n * 3 + 2];

    // Gather neighbor k for this point.
    const long long idx64 = neigh_idx[((size_t)(bq * N_PTS + n)) * K_NB + k];
    const int idx = (int)idx64;
    const float nx = xb[idx * 3 + 0];
    const float ny = xb[idx * 3 + 1];
    const float nz = xb[idx * 3 + 2];

    const float rx = tx - nx, ry = ty - ny, rz = tz - nz;
    const float dis = __builtin_amdgcn_sqrtf(rx * rx + ry * ry + rz * rz);

    // B operand (4x16 f32, K x N): VGPR v holds row (K%2==v); lane half holds
    // K-pair 0-1 (lanes 0-15) or 2-3 (lanes 16-31); N = neighbor k = lane&15.
    // Channels: 0=dis 1=rx 2=ry 3=rz | 4=tx 5=ty 6=tz 7=nx | 8=ny 9=nz 10,11=pad
    v2f b0, b1, b2;
    b0[0] = hi ? ry : dis;  b0[1] = hi ? rz  : rx;
    b1[0] = hi ? tz : tx;   b1[1] = hi ? nx  : ty;
    b2[0] = hi ? 0.f : ny;  b2[1] = hi ? 0.f : nz;

    // A operand (16x4 f32, M x K): M = output channel = lane&15, VGPR v holds
    // K%2==v, lane half selects K-pair. Chunk j covers channels 4j..4j+3.
    const float* Wm = Wmat + k * 10;   // row m = lane&15
    const int cb = hi ? 2 : 0;
    v2f a0, a1, a2;
    a0[0] = Wm[cb + 0];  a0[1] = Wm[cb + 1];
    a1[0] = Wm[cb + 4];  a1[1] = Wm[cb + 5];
    a2[0] = hi ? 0.f : Wm[8];
    a2[1] = hi ? 0.f : Wm[9];

    v8f acc = {};
    acc = __builtin_amdgcn_wmma_f32_16x16x4_f32(false, a0, false, b0,
                                                (short)0, acc, false, false);
    acc = __builtin_amdgcn_wmma_f32_16x16x4_f32(false, a1, false, b1,
                                                (short)0, acc, false, false);
    acc = __builtin_amdgcn_wmma_f32_16x16x4_f32(false, a2, false, b2,
                                                (short)0, acc, false, false);

    // D layout: VGPR r holds output channel o = r + (hi ? 8 : 0), column k.
    // Fused eval-mode BN + ReLU, then streaming (non-temporal) stores:
    // each (r, half-wave) writes 16 consecutive floats (64B burst).
    const int obase = hi ? 8 : 0;
    float* outb = out + (((size_t)bq * D_OUT) * N_PTS + n) * K_NB + k;
#pragma unroll
    for (int r = 0; r < 8; ++r) {
        const int o = obase + r;
        const float g   = gamma[o];
        const float inv = __builtin_amdgcn_rsqf(rvar[o] + BN_EPS);
        const float sc  = g * inv;
        const float sh  = beta[o] - g * rmean[o] * inv;
        float y = fmaf(acc[r], sc, sh);
        y = fmaxf(y, 0.f);
        __builtin_nontemporal_store(y, outb + (size_t)o * (N_PTS * K_NB));
    }
}

extern "C" void kernel_launch(void* const* d_in, const int* in_sizes, int n_in,
                              void* d_out, int out_size, void* d_ws, size_t ws_size,
                              hipStream_t stream) {
    (void)in_sizes; (void)n_in; (void)out_size; (void)d_ws; (void)ws_size;
    const float*     xyz   = (const float*)d_in[0];
    /* d_in[1] = feature: unused by the reference computation */
    const long long* nidx  = (const long long*)d_in[2];
    const float*     Wmat  = (const float*)d_in[3];
    const float*     gamma = (const float*)d_in[4];
    const float*     beta  = (const float*)d_in[5];
    const float*     rmean = (const float*)d_in[6];
    const float*     rvar  = (const float*)d_in[7];
    float* out = (float*)d_out;

    const int total_waves = B_SZ * N_PTS;          // 163840
    const int blocks      = total_waves / 8;       // 8 waves (256 thr) per block
    relpos_enc_wmma_kernel<<<blocks, 256, 0, stream>>>(
        xyz, nidx, Wmat, gamma, beta, rmean, rvar, out);
}